// Lossv9_39256001085460
// MI455X (gfx1250) — compile-verified
//
#include <hip/hip_runtime.h>

typedef __attribute__((ext_vector_type(2))) float v2f;
typedef __attribute__((ext_vector_type(8))) float v8f;

// Problem geometry (fixed by the reference).
constexpr int Bn   = 4;
constexpr int Hh   = 512;
constexpr int Ww   = 512;
constexpr int HW   = Hh * Ww;        // 262144 = 2^18
constexpr int CHW  = 3 * HW;         // per-(view,batch) stride
constexpr int VSTR = Bn * CHW;       // per-view stride in a [9,B,3,H,W] stack
constexpr int NPIX = Bn * HW;        // 1,048,576 pixels
constexpr int BLK  = 256;            // 8 waves (wave32)
constexpr int NBLK = NPIX / BLK;     // 4096 partial sums

__global__ __launch_bounds__(BLK) void lf_loss_main(
    const float* __restrict__ disp,
    const float* __restrict__ vx,  const float* __restrict__ vy,
    const float* __restrict__ v45, const float* __restrict__ v135,
    float* __restrict__ partial)
{
    const int t   = blockIdx.x * BLK + threadIdx.x;
    const int b   = t >> 18;            // / HW
    const int rem = t & (HW - 1);
    const int i   = rem >> 9;           // / W
    const int j   = rem & (Ww - 1);

    const float d = disp[t];            // pred_disp[:,0] flat == [B,H,W] flat

    const float* views[4] = { vx, vy, v45, v135 };
    const float  su[4]    = { 0.f, 1.f, 1.f,  1.f };   // du scale (k-4)
    const float  sv[4]    = { 1.f, 0.f, 1.f, -1.f };   // dv scale (k-4)

    const float luma[3] = { 0.299f, 0.587f, 0.114f };

    float acc = 0.f;

    #pragma unroll
    for (int dir = 0; dir < 4; ++dir) {
        const int pix = i * Ww + j;
        // Warm L2 for the first non-central view of this direction (CDNA5
        // global_prefetch_b8; addresses are valid -> safe speculative RT).
        __builtin_prefetch(views[dir] + b * CHW + pix, 0, 1);

        // Central view (k=4): zero offset -> exact pixel, no interpolation.
        const float* c4 = views[dir] + 4 * VSTR + b * CHW + pix;
        const float cc[3] = { c4[0], c4[HW], c4[2 * HW] };

        float maps[9];
        maps[4] = 0.f;
        #pragma unroll
        for (int k = 0; k < 9; ++k) {
            if (k == 4) continue;
            const float f  = (float)(k - 4);
            float px = (float)j - d * (sv[dir] * f);
            float py = (float)i - d * (su[dir] * f);
            px = fminf(fmaxf(px, 0.f), (float)(Ww - 1));
            py = fminf(fmaxf(py, 0.f), (float)(Hh - 1));
            const float x0f = floorf(px), y0f = floorf(py);
            const int   x0  = (int)x0f,  y0  = (int)y0f;
            const int   x1  = (x0 < Ww - 1) ? x0 + 1 : Ww - 1;
            const int   y1  = (y0 < Hh - 1) ? y0 + 1 : Hh - 1;
            const float wx = px - x0f, wy = py - y0f;
            const float w00 = (1.f - wy) * (1.f - wx);
            const float w01 = (1.f - wy) * wx;
            const float w10 = wy * (1.f - wx);
            const float w11 = wy * wx;
            const int o00 = y0 * Ww + x0, o01 = y0 * Ww + x1;
            const int o10 = y1 * Ww + x0, o11 = y1 * Ww + x1;

            const float* vb = views[dir] + k * VSTR + b * CHW;
            float m = 0.f;
            #pragma unroll
            for (int c = 0; c < 3; ++c) {
                const float* p = vb + c * HW;
                const float val = p[o00] * w00 + p[o01] * w01
                                + p[o10] * w10 + p[o11] * w11;
                m += luma[c] * fabsf(val - cc[c]);
            }
            maps[k] = m;
        }

        // Prefix sums and windowed averages (d[0..8]).
        float cs[9]; float run = 0.f;
        #pragma unroll
        for (int k = 0; k < 9; ++k) { run += maps[k]; cs[k] = run; }
        const float total = cs[8];
        float dv_[9];
        dv_[0] = total / 9.f;
        dv_[1] = (total - cs[0]) / 8.f;
        dv_[2] = (total - cs[1]) / 7.f;
        dv_[3] = (total - cs[2]) / 6.f;
        dv_[4] = (total - cs[3]) / 5.f;
        dv_[5] = cs[7] / 8.f;
        dv_[6] = cs[6] / 7.f;
        dv_[7] = cs[5] / 6.f;
        dv_[8] = cs[4] / 5.f;
        float dmin = dv_[0];
        #pragma unroll
        for (int k = 1; k < 9; ++k) dmin = fminf(dmin, dv_[k]);
        const bool occ = fabsf(dv_[8] - dv_[4]) < 0.01f;
        acc += occ ? dv_[0] : dmin;
    }

    acc *= (1.0f / (float)NPIX);   // the "mean" factor, identical per direction

    // Deterministic block reduction: wave32 shuffle tree + LDS across 8 waves.
    #pragma unroll
    for (int off = 16; off > 0; off >>= 1) acc += __shfl_down(acc, off, 32);
    __shared__ float sred[BLK / 32];
    const int lane = threadIdx.x & 31, wid = threadIdx.x >> 5;
    if (lane == 0) sred[wid] = acc;
    __syncthreads();
    if (threadIdx.x == 0) {
        float s = 0.f;
        #pragma unroll
        for (int w = 0; w < BLK / 32; ++w) s += sred[w];
        partial[blockIdx.x] = s;
    }
}

// Final 4096 -> 1 reduction. The 64 per-lane partial sums of each wave are
// packed as a 16x4 F32 A-matrix; D = A x ones via V_WMMA_F32_16X16X4_F32 puts
// the row-sums in every D column, so (column sum) + (xor-16 partner) is the
// exact wave total. 8 wave totals are folded in fixed order by thread 0.
__global__ __launch_bounds__(BLK) void lf_loss_reduce(
    const float* __restrict__ partial, float* __restrict__ out)
{
    const int tid = threadIdx.x;
    float s0 = 0.f, s1 = 0.f;
    const int base = tid * (NBLK / BLK);          // 16 partials per thread
    #pragma unroll
    for (int q = 0; q < 8; ++q) s0 += partial[base + q];
    #pragma unroll
    for (int q = 0; q < 8; ++q) s1 += partial[base + 8 + q];

    v2f a;  a.x = s0;  a.y = s1;                  // A: 16x4 f32 (2 VGPRs/lane)
    v2f bo; bo.x = 1.f; bo.y = 1.f;               // B: all-ones 4x16
    v8f c = {};
    c = __builtin_amdgcn_wmma_f32_16x16x4_f32(
            /*neg_a=*/false, a, /*neg_b=*/false, bo,
            /*c_mod=*/(short)0, c, /*reuse_a=*/false, /*reuse_b=*/false);
    float col = c[0] + c[1] + c[2] + c[3] + c[4] + c[5] + c[6] + c[7];
    const float wavesum = col + __shfl_xor(col, 16, 32);

    __shared__ float sred[BLK / 32];
    const int lane = tid & 31, wid = tid >> 5;
    if (lane == 0) sred[wid] = wavesum;
    __syncthreads();
    if (tid == 0) {
        float s = 0.f;
        #pragma unroll
        for (int w = 0; w < BLK / 32; ++w) s += sred[w];
        out[0] = s;
    }
}

extern "C" void kernel_launch(void* const* d_in, const int* in_sizes, int n_in,
                              void* d_out, int out_size, void* d_ws, size_t ws_size,
                              hipStream_t stream)
{
    const float* disp = (const float*)d_in[0];   // pred_disp [4,1,512,512]
    const float* vx   = (const float*)d_in[1];   // views_x   [9,4,3,512,512]
    const float* vy   = (const float*)d_in[2];   // views_y
    const float* v45  = (const float*)d_in[3];   // views_45
    const float* v135 = (const float*)d_in[4];   // views_135
    float* partial = (float*)d_ws;               // 4096 floats of scratch
    float* outp    = (float*)d_out;              // 1 float

    lf_loss_main<<<NBLK, BLK, 0, stream>>>(disp, vx, vy, v45, v135, partial);
    lf_loss_reduce<<<1, BLK, 0, stream>>>(partial, outp);
}